// QuantumLayer_65189013619085
// MI455X (gfx1250) — compile-verified
//
#include <hip/hip_runtime.h>
#include <hip/hip_bf16.h>

typedef __attribute__((ext_vector_type(2))) float v2f;
typedef __attribute__((ext_vector_type(8))) float v8f;

// ---------------------------------------------------------------------------
// Kernel 1: fold 5 layers of shared U3 gates + ring CNOTs into a single 16x16
// complex unitary U.  Thread t owns column t of U.  Output is U^T stored as
// two real planes: bt[t*16+s] = Re U[s][t], bt[256+t*16+s] = Im U[s][t].
// ---------------------------------------------------------------------------
__global__ void build_unitary(const float* __restrict__ w, float* __restrict__ bt) {
    int t = threadIdx.x;
    if (t >= 16) return;
    float cre[16], cim[16];
#pragma unroll
    for (int s = 0; s < 16; ++s) { cre[s] = (s == t) ? 1.0f : 0.0f; cim[s] = 0.0f; }

    for (int l = 0; l < 5; ++l) {
#pragma unroll
        for (int i = 0; i < 4; ++i) {
            float th = w[(l * 4 + i) * 3 + 0];
            float ph = w[(l * 4 + i) * 3 + 1];
            float la = w[(l * 4 + i) * 3 + 2];
            float st, ct, sp, cp, sl, cl;
            __sincosf(th * 0.5f, &st, &ct);
            __sincosf(ph, &sp, &cp);
            __sincosf(la, &sl, &cl);
            // U3 = [[ct, -e^{i la} st], [e^{i ph} st, e^{i(ph+la)} ct]]
            float g00r = ct,        g00i = 0.0f;
            float g01r = -cl * st,  g01i = -sl * st;
            float g10r = cp * st,   g10i = sp * st;
            float er   = cp * cl - sp * sl;
            float ei   = cp * sl + sp * cl;
            float g11r = er * ct,   g11i = ei * ct;
            const int mask = 1 << (3 - i);
#pragma unroll
            for (int r0 = 0; r0 < 16; ++r0) {
                if (r0 & mask) continue;
                int r1 = r0 | mask;
                float ar = cre[r0], ai = cim[r0], br = cre[r1], bi = cim[r1];
                cre[r0] = g00r * ar - g00i * ai + g01r * br - g01i * bi;
                cim[r0] = g00r * ai + g00i * ar + g01r * bi + g01i * br;
                cre[r1] = g10r * ar - g10i * ai + g11r * br - g11i * bi;
                cim[r1] = g10r * ai + g10i * ar + g11r * bi + g11i * br;
            }
        }
        const int cw[4] = {0, 1, 2, 3}, tw[4] = {1, 2, 3, 0};
#pragma unroll
        for (int e = 0; e < 4; ++e) {
            const int mc = 1 << (3 - cw[e]);
            const int mt = 1 << (3 - tw[e]);
#pragma unroll
            for (int r = 0; r < 16; ++r) {
                if ((r & mc) && !(r & mt)) {
                    int r2 = r | mt;
                    float tr = cre[r], ti = cim[r];
                    cre[r] = cre[r2]; cim[r] = cim[r2];
                    cre[r2] = tr;     cim[r2] = ti;
                }
            }
        }
    }
#pragma unroll
    for (int s = 0; s < 16; ++s) {
        bt[t * 16 + s]       = cre[s];
        bt[256 + t * 16 + s] = cim[s];
    }
}

// ---------------------------------------------------------------------------
// Kernel 2: one wave per 16-sample tile.  Kronecker-structured psi0 (no big
// register arrays, no dynamic indexing) -> 16x v_wmma_f32_16x16x4_f32 complex
// GEMM -> |psi|^2 -> folded Walsh Z-sums via LDS -> float4 store.
// ---------------------------------------------------------------------------
__global__ __launch_bounds__(256) void qnn_kernel(const float* __restrict__ x,
                                                  const float* __restrict__ bt,
                                                  float* __restrict__ out,
                                                  int nsamp) {
    __shared__ float lds[8 * 16 * 16];
    const int lane = threadIdx.x & 31;
    const int wave = threadIdx.x >> 5;
    const int m    = lane & 15;   // sample row within tile / B column
    const int h    = lane >> 4;   // half-wave selector (K sub-pair / wire-2 bit)
    const int tile = blockIdx.x * 8 + wave;
    if (tile * 16 >= nsamp) return;          // wave-uniform: EXEC stays full
    int sample = tile * 16 + m;
    if (sample >= nsamp) sample = nsamp - 1;

    // --- per-wire factors: only sincos(x/2); derive sin/cos(x) from it
    const float4 xs = reinterpret_cast<const float4*>(x)[sample];
    const float xi[4] = {xs.x, xs.y, xs.z, xs.w};
    float c0[4], v1r[4], v1i[4];
#pragma unroll
    for (int i = 0; i < 4; ++i) {
        float st, ct;
        __sincosf(xi[i] * 0.5f, &st, &ct);
        float sx = 2.0f * st * ct;        // sin(x)
        float cx = 1.0f - 2.0f * st * st; // cos(x)
        c0[i]  = ct;                      // <0| branch (real)
        v1r[i] = cx * st;                 // <1| branch = e^{ix} sin(x/2)
        v1i[i] = sx * st;
    }

    // --- Kronecker halves: q[hi2] over wires 0,1 (all four, constant-indexed)
    float qr[4], qi[4];
    qr[0] = c0[0] * c0[1];                          qi[0] = 0.0f;
    qr[1] = c0[0] * v1r[1];                         qi[1] = c0[0] * v1i[1];
    qr[2] = v1r[0] * c0[1];                         qi[2] = v1i[0] * c0[1];
    qr[3] = v1r[0] * v1r[1] - v1i[0] * v1i[1];      qi[3] = v1r[0] * v1i[1] + v1i[0] * v1r[1];

    // --- wires 2,3: this lane only touches lo2 in {2h, 2h+1}; wire-2 bit == h
    float f2r = h ? v1r[2] : c0[2];
    float f2i = h ? v1i[2] : 0.0f;
    float rAr = f2r * c0[3];                        // lo2 = 2h   (wire3 = 0)
    float rAi = f2i * c0[3];
    float rBr = f2r * v1r[3] - f2i * v1i[3];        // lo2 = 2h+1 (wire3 = 1)
    float rBi = f2r * v1i[3] + f2i * v1r[3];

    // --- complex GEMM: psi = psi0 * U^T, K=16 as 4 chunks of 16x16x4 fp32 WMMA
    const float* __restrict__ btre = bt;
    const float* __restrict__ btim = bt + 256;
    v8f accr = {};
    v8f acci = {};
#pragma unroll
    for (int c = 0; c < 4; ++c) {
        // A: lanes 0-15 hold K = 4c+0,4c+1 ; lanes 16-31 hold K = 4c+2,4c+3
        v2f ar, ai;
        ar.x = qr[c] * rAr - qi[c] * rAi;
        ai.x = qr[c] * rAi + qi[c] * rAr;
        ar.y = qr[c] * rBr - qi[c] * rBi;
        ai.y = qr[c] * rBi + qi[c] * rBr;
        // B: vgpr v, half h -> row K = 4c + 2h + v, column n = lane%16
        const int k = 4 * c + 2 * h;
        v2f br, bi, nbi;
        br.x = btre[k * 16 + m];        br.y = btre[(k + 1) * 16 + m];
        bi.x = btim[k * 16 + m];        bi.y = btim[(k + 1) * 16 + m];
        nbi.x = -bi.x;                  nbi.y = -bi.y;
        // Re += Ar*Br - Ai*Bi   (explicit -Bi: F32 WMMA NEG bits are C-only)
        accr = __builtin_amdgcn_wmma_f32_16x16x4_f32(false, ar, false, br,  (short)0, accr, false, false);
        accr = __builtin_amdgcn_wmma_f32_16x16x4_f32(false, ai, false, nbi, (short)0, accr, false, false);
        // Im += Ar*Bi + Ai*Br
        acci = __builtin_amdgcn_wmma_f32_16x16x4_f32(false, ar, false, bi,  (short)0, acci, false, false);
        acci = __builtin_amdgcn_wmma_f32_16x16x4_f32(false, ai, false, br,  (short)0, acci, false, false);
    }

    // --- probabilities to LDS: row M = v + 8h (sample), col N = m (state)
    float* wl = lds + wave * 256;
#pragma unroll
    for (int v = 0; v < 8; ++v) {
        float p = accr[v] * accr[v] + acci[v] * acci[v];
        wl[(v + 8 * h) * 16 + m] = p;
    }
    __syncthreads();

    // --- folded Walsh reduction: z_i = sum_s (-1)^{bit_{3-i}(s)} p[s]
    if (h == 0) {
        const float* pr = wl + m * 16;
        float e[8], z3 = 0.0f;
#pragma unroll
        for (int t = 0; t < 8; ++t) {
            float a = pr[2 * t], b = pr[2 * t + 1];
            e[t] = a + b;
            z3  += a - b;
        }
        float e2[4], z2 = 0.0f;
#pragma unroll
        for (int u = 0; u < 4; ++u) {
            z2   += e[2 * u] - e[2 * u + 1];
            e2[u] = e[2 * u] + e[2 * u + 1];
        }
        float z1 = (e2[0] - e2[1]) + (e2[2] - e2[3]);
        float z0 = (e2[0] + e2[1]) - (e2[2] + e2[3]);
        float4 o = {z0, z1, z2, z3};
        reinterpret_cast<float4*>(out)[sample] = o;
    }
}

extern "C" void kernel_launch(void* const* d_in, const int* in_sizes, int n_in,
                              void* d_out, int out_size, void* d_ws, size_t ws_size,
                              hipStream_t stream) {
    const float* x = (const float*)d_in[0];   // [B,4] fp32
    const float* w = (const float*)d_in[1];   // [5,4,3] fp32
    float* out = (float*)d_out;               // [B,4] fp32
    float* bt  = (float*)d_ws;                // 512 floats: U^T re/im planes
    const int nsamp = in_sizes[0] / 4;

    hipLaunchKernelGGL(build_unitary, dim3(1), dim3(16), 0, stream, w, bt);

    const int tiles  = (nsamp + 15) / 16;
    const int blocks = (tiles + 7) / 8;
    hipLaunchKernelGGL(qnn_kernel, dim3(blocks), dim3(256), 0, stream, x, bt, out, nsamp);
}